// NDFT_86474871537949
// MI455X (gfx1250) — compile-verified
//
#include <hip/hip_runtime.h>

typedef __attribute__((ext_vector_type(2))) float v2f;
typedef __attribute__((ext_vector_type(8))) float v8f;

#define M_PTS 8192
#define NGRID 64                 // 64x64 frequency grid
#define ROWS_PER_BLOCK 128       // 8 waves * 16 rows
#define TWO_PI_F 6.28318530717958647692f

// y[b,m] = sum_{jx,jy} f[b,jx,jy] * exp(-2pi i ((jx-32)*x0 + (jy-32)*x1))
// Factor:  E[m,jx] = exp(-2pi i (jx-32) x0[m])   (complex, 16x64 per wave)
//          T = E @ F  (two real fp32 GEMMs via V_WMMA_F32_16X16X4_F32)
//          y[m] = sum_jy T[m,jy] * exp(-2pi i (jy-32) x1[m])
__launch_bounds__(256)
__global__ void ndft_sep_wmma_kernel(const float* __restrict__ x,
                                     const float* __restrict__ fhat,
                                     float* __restrict__ out)
{
    __shared__ float sF[NGRID * NGRID];     // f_hat[b] row-major [jx][jy], 16 KB
    __shared__ float sX0[ROWS_PER_BLOCK];
    __shared__ float sX1[ROWS_PER_BLOCK];

    const int b     = blockIdx.x >> 6;      // 64 m-tiles per batch
    const int m0    = (blockIdx.x & 63) * ROWS_PER_BLOCK;
    const int tid   = threadIdx.x;

    // Stage f_hat[b] (4096 f32) and the x tile into LDS.
    const float* Fb = fhat + (size_t)b * NGRID * NGRID;
    #pragma unroll
    for (int i = 0; i < (NGRID * NGRID) / 256; ++i)
        sF[tid + i * 256] = Fb[tid + i * 256];
    if (tid < ROWS_PER_BLOCK) {
        const float* xb = x + ((size_t)b * M_PTS + m0 + tid) * 2;
        sX0[tid] = xb[0];
        sX1[tid] = xb[1];
    }
    __syncthreads();

    const int wave     = tid >> 5;          // 0..7
    const int lane     = tid & 31;
    const int halflane = lane & 15;
    const bool hi      = (lane >= 16);
    const int rowbase  = wave * 16;         // 16 M-rows per wave

    // A-operand row for this lane (lanes 0-15 and 16-31 both cover M=0..15).
    const float x0 = sX0[rowbase + halflane];

    v8f acc_re[4], acc_im[4];
    #pragma unroll
    for (int t = 0; t < 4; ++t) { acc_re[t] = (v8f){}; acc_im[t] = (v8f){}; }

    // K-chain: 16 steps of K=4 over the 64 jx frequencies.
    #pragma unroll 4
    for (int k0 = 0; k0 < NGRID; k0 += 4) {
        const int j0 = k0 + (hi ? 2 : 0);   // this lane's K pair: j0, j0+1
        float s0, c0, s1, c1;
        __sincosf(TWO_PI_F * (float)(j0 - 32) * x0, &s0, &c0);
        __sincosf(TWO_PI_F * (float)(j0 - 31) * x0, &s1, &c1);
        v2f a_re; a_re.x = c0;  a_re.y = c1;
        v2f a_im; a_im.x = -s0; a_im.y = -s1;

        #pragma unroll
        for (int t = 0; t < 4; ++t) {
            // B-operand (4x16 tile of F): lane holds rows j0, j0+1 at column t*16+halflane
            v2f bm;
            bm.x = sF[(size_t)j0 * NGRID + t * 16 + halflane];
            bm.y = sF[(size_t)(j0 + 1) * NGRID + t * 16 + halflane];
            acc_re[t] = __builtin_amdgcn_wmma_f32_16x16x4_f32(
                false, a_re, false, bm, (short)0, acc_re[t], false, false);
            acc_im[t] = __builtin_amdgcn_wmma_f32_16x16x4_f32(
                false, a_im, false, bm, (short)0, acc_im[t], false, false);
        }
    }

    // Combine: y[m] = sum_jy (Tre + i Tim) * (cos - i sin), theta = 2pi (jy-32) x1[m].
    // C/D layout: component r -> row r (lanes 0-15) or row r+8 (lanes 16-31),
    // column = t*16 + halflane.
    #pragma unroll
    for (int r = 0; r < 8; ++r) {
        const int mrow = rowbase + (hi ? 8 : 0) + r;
        const float x1 = sX1[mrow];
        float re = 0.f, im = 0.f;
        #pragma unroll
        for (int t = 0; t < 4; ++t) {
            const int j = t * 16 + halflane;
            float s, c;
            __sincosf(TWO_PI_F * (float)(j - 32) * x1, &s, &c);
            const float tr = acc_re[t][r];
            const float ti = acc_im[t][r];
            re += tr * c + ti * s;
            im += ti * c - tr * s;
        }
        // Reduce across the 16 lanes of this half-wave (wave32).
        #pragma unroll
        for (int off = 8; off >= 1; off >>= 1) {
            re += __shfl_xor(re, off, 16);
            im += __shfl_xor(im, off, 16);
        }
        if (halflane == 0) {
            const size_t gm = (size_t)b * M_PTS + (m0 + mrow);
            out[gm * 2 + 0] = re;
            out[gm * 2 + 1] = im;
        }
    }
}

extern "C" void kernel_launch(void* const* d_in, const int* in_sizes, int n_in,
                              void* d_out, int out_size, void* d_ws, size_t ws_size,
                              hipStream_t stream) {
    (void)in_sizes; (void)n_in; (void)out_size; (void)d_ws; (void)ws_size;
    const float* x    = (const float*)d_in[0];   // [B, M, 2] f32
    const float* fhat = (const float*)d_in[1];   // [B, 64, 64] f32
    float* out        = (float*)d_out;           // [B, M] complex64 -> interleaved f32

    const int blocks = 2 * (M_PTS / ROWS_PER_BLOCK);   // B * 64 = 128
    ndft_sep_wmma_kernel<<<blocks, 256, 0, stream>>>(x, fhat, out);
}